// StateManager_34385508171799
// MI455X (gfx1250) — compile-verified
//
#include <hip/hip_runtime.h>
#include <cstdint>
#include <cstddef>

typedef __attribute__((ext_vector_type(2))) float v2f;
typedef __attribute__((ext_vector_type(8))) float v8f;

#define K_DIM   256
#define V_DIM   256
#define VQ      64          // columns handled per block (quarter of V)
#define THREADS 128         // 4 waves of 32
#define CLIPV   1e6f

// LDS budget: 256*64*4 = 64KB state tile + 3*1KB (keys/gates/kg) + 2*256B (v, kt)
// = 69120 B -> ~4 blocks per WGP (320KB LDS).

__global__ __launch_bounds__(THREADS)
void kda_state_update(const float* __restrict__ state,
                      const float* __restrict__ keys,
                      const float* __restrict__ values,
                      const float* __restrict__ gates,
                      const float* __restrict__ beta,
                      float* __restrict__ out)
{
    extern __shared__ float smem[];
    float* sS    = smem;                 // [256][64] state tile (original values)
    float* sKeys = smem + K_DIM * VQ;    // [256]
    float* sG    = sKeys + K_DIM;        // [256]
    float* sKg   = sG    + K_DIM;        // [256] keys*gates
    float* sV    = sKg   + K_DIM;        // [64]
    float* sKt   = sV    + VQ;           // [64]  k^T (Diag(a) S) for this quarter

    const int head = blockIdx.x >> 2;    // 1024 heads
    const int q    = blockIdx.x & 3;     // V quarter
    const int tid  = threadIdx.x;

    const size_t headKV = (size_t)head * (size_t)(K_DIM * V_DIM);
    const float* sbase  = state + headKV;          // uniform -> SGPR pair

    // ---------------- Phase 0a: small vectors into LDS ----------------
    for (int i = tid; i < K_DIM; i += THREADS) {
        float k = keys [head * K_DIM + i];
        float g = gates[head * K_DIM + i];
        sKeys[i] = k;
        sG[i]    = g;
        sKg[i]   = k * g;
    }
    if (tid < VQ) sV[tid] = values[head * V_DIM + q * VQ + tid];

    // ---------------- Phase 0b: async DMA state quarter -> LDS ----------------
    // 256 rows x 64 cols x 4B = 4096 chunks of 16B; 16 chunks per row.
    {
        const uint32_t ldsBase  = (uint32_t)(size_t)(void*)sS;       // generic->LDS addr (low 32b)
        const uint64_t base64   = (uint64_t)(size_t)sbase;
        const int      qbyte    = q * VQ * 4;
        for (int c = tid; c < (K_DIM * VQ / 4); c += THREADS) {
            int row = c >> 4;
            int c16 = c & 15;
            uint32_t voff  = (uint32_t)(row * (V_DIM * 4) + qbyte + c16 * 16);
            uint32_t laddr = ldsBase + (uint32_t)(row * (VQ * 4) + c16 * 16);
            asm volatile("global_load_async_to_lds_b128 %0, %1, %2"
                         :: "v"(laddr), "v"(voff), "s"(base64) : "memory");
        }
        asm volatile("s_wait_asynccnt 0" ::: "memory");
    }
    __syncthreads();

    // ---------------- Phase 1: kt[v] = sum_k kg[k]*S[k][v] via WMMA f32 16x16x4 ----
    // A[m][j] = S[k0+j][v0+m]  (M-dim = v), B[j][n] = kg[k0+j] (broadcast over n)
    // => C[m][n] accumulates kt[v0+m] in every column n.
    {
        const int wave = tid >> 5;       // 0..3 -> v tile
        const int lane = tid & 31;
        const int half = lane >> 4;      // 0: K=0,1   1: K=2,3
        const int lm   = lane & 15;
        const int v0   = wave * 16;

        v8f c = {0.f, 0.f, 0.f, 0.f, 0.f, 0.f, 0.f, 0.f};
        for (int k0 = 0; k0 < K_DIM; k0 += 4) {
            const int kb = k0 + 2 * half;
            v2f a, b;
            a.x = sS[kb       * VQ + v0 + lm];   // A[lm][2*half+0]
            a.y = sS[(kb + 1) * VQ + v0 + lm];   // A[lm][2*half+1]
            b.x = sKg[kb];                       // B[2*half+0][*]
            b.y = sKg[kb + 1];                   // B[2*half+1][*]
            c = __builtin_amdgcn_wmma_f32_16x16x4_f32(
                    false, a, false, b, (short)0, c, false, false);
        }
        // C layout: vgpr r -> lanes0-15: M=r, lanes16-31: M=r+8
        if (lane == 0) {
            #pragma unroll
            for (int r = 0; r < 8; ++r) sKt[v0 + r] = c[r];
        } else if (lane == 16) {
            #pragma unroll
            for (int r = 0; r < 8; ++r) sKt[v0 + 8 + r] = c[r];
        }
    }
    __syncthreads();

    // ---------------- Phase 2: S' = g[k]*S + keys[k]*beta*(v - kt), clamp/NaN, store ----
    {
        const int v4l    = (tid & 15) * 4;   // 16 col-groups of 4
        const int rowoff = tid >> 4;         // 0..7 rows per sweep
        const float4 kt4  = *(const float4*)&sKt[v4l];
        const float4 val4 = *(const float4*)&sV[v4l];
        const float  bta  = beta[head];
        const float  c0 = bta * (val4.x - kt4.x);
        const float  c1 = bta * (val4.y - kt4.y);
        const float  c2 = bta * (val4.z - kt4.z);
        const float  c3 = bta * (val4.w - kt4.w);

        float* obase = out + headKV + (size_t)(q * VQ);
        for (int kb = 0; kb < K_DIM; kb += 8) {
            const int   k  = kb + rowoff;
            const float g  = sG[k];
            const float ky = sKeys[k];
            const float4 s = *(const float4*)&sS[k * VQ + v4l];
            float4 r;
            r.x = fmaf(ky, c0, g * s.x);
            r.y = fmaf(ky, c1, g * s.y);
            r.z = fmaf(ky, c2, g * s.z);
            r.w = fmaf(ky, c3, g * s.w);
            // elementwise Inf clip to +-1e6; NaN -> revert to previous state
            r.x = (r.x != r.x) ? s.x : fminf(fmaxf(r.x, -CLIPV), CLIPV);
            r.y = (r.y != r.y) ? s.y : fminf(fmaxf(r.y, -CLIPV), CLIPV);
            r.z = (r.z != r.z) ? s.z : fminf(fmaxf(r.z, -CLIPV), CLIPV);
            r.w = (r.w != r.w) ? s.w : fminf(fmaxf(r.w, -CLIPV), CLIPV);
            *(float4*)(obase + (size_t)k * V_DIM + v4l) = r;
        }
    }
}

extern "C" void kernel_launch(void* const* d_in, const int* in_sizes, int n_in,
                              void* d_out, int out_size, void* d_ws, size_t ws_size,
                              hipStream_t stream) {
    (void)in_sizes; (void)n_in; (void)out_size; (void)d_ws; (void)ws_size;
    const float* state  = (const float*)d_in[0];
    const float* keys   = (const float*)d_in[1];
    const float* values = (const float*)d_in[2];
    const float* gates  = (const float*)d_in[3];
    const float* beta   = (const float*)d_in[4];
    float* out = (float*)d_out;

    const int blocks = 32 * 32 * 4;  // B*H heads x 4 V-quarters
    const size_t shmem = (size_t)(K_DIM * VQ + 3 * K_DIM + 2 * VQ) * sizeof(float);
    kda_state_update<<<blocks, THREADS, shmem, stream>>>(state, keys, values, gates, beta, out);
}